// PIGNet_89627377533532
// MI455X (gfx1250) — compile-verified
//
#include <hip/hip_runtime.h>

// ---- static problem config (matches reference) ----
#define NBATCH 96
#define NLIG   30
#define NPOC   200
#define NPER   230
#define NN     (NBATCH * NPER)        // 22080 nodes
#define DGD    128
#define FIND   54
#define NE1    (NN * 8)               // 176640 intra edges
#define NE2    (NBATCH * 2000)        // 192000 conv edges
#define NE3    (NBATCH * NLIG * NPOC) // 576000 interaction edges

typedef float v2f __attribute__((ext_vector_type(2)));
typedef float v8f __attribute__((ext_vector_type(8)));

// ---------- helpers ----------
__device__ __forceinline__ unsigned fkey(float f) {
  unsigned u = __float_as_uint(f);
  return (u & 0x80000000u) ? ~u : (u | 0x80000000u);
}
__device__ __forceinline__ float funkey(unsigned k) {
  unsigned u = (k & 0x80000000u) ? (k & 0x7fffffffu) : ~k;
  return __uint_as_float(u);
}
__device__ __forceinline__ float wred(float v) {
#pragma unroll
  for (int off = 16; off > 0; off >>= 1) v += __shfl_xor(v, off, 32);
  return v;
}

// ---------- f32 WMMA GEMM: C[M x 128] = A[M x K] @ W[K x 128] (+bias)(relu)(+resid) ----------
// W staged in LDS with rows k / k+1 interleaved per column:
//   sW[(k>>1)*PSTR + 2*c + (k&1)]
// so each B fragment (rows k,k+1 at column c) is ONE contiguous ds_load_b64.
// PSTR = 288 floats (pad 32): the odd row-pair read by lanes 16-31 is shifted by
// exactly 32 banks vs the even pair read by lanes 0-15 -> conflict-free b64 reads.
// Each wave owns 32 output rows: 2 A fragments, 16 v8f accumulators, 16 WMMAs/k-step.
// flags: 1 = add bias, 2 = relu, 4 = add residual
template <int K>
__global__ __launch_bounds__(256) void k_gemm_wmma(
    const float* __restrict__ A, const float* __restrict__ W,
    const float* __restrict__ bias, const float* __restrict__ resid,
    float* __restrict__ C, int M, int flags) {
  constexpr int KP    = (K + 3) & ~3;   // padded row count (56 or 128)
  constexpr int NPAIR = KP / 2;         // row pairs (28 or 64)
  constexpr int PSTR  = 288;            // floats per row-pair (256 data + 32 pad)
  __shared__ float sW[NPAIR * PSTR];    // 32.3 KB / 73.7 KB (CDNA5: 320 KB/WG)

  // cooperative stage: interleave row pairs of W[K x 128] into LDS (zero-pad rows >= K)
  for (int idx = threadIdx.x; idx < NPAIR * 32; idx += 256) {
    const int p  = idx >> 5;
    const int c4 = (idx & 31) << 2;
    const int kr0 = 2 * p, kr1 = kr0 + 1;
    float4 r0 = make_float4(0.f, 0.f, 0.f, 0.f);
    float4 r1 = make_float4(0.f, 0.f, 0.f, 0.f);
    if (kr0 < K) r0 = *(const float4*)(W + (size_t)kr0 * DGD + c4);
    if (kr1 < K) r1 = *(const float4*)(W + (size_t)kr1 * DGD + c4);
    float* dp = sW + p * PSTR + c4 * 2;
    *(float4*)(dp)     = make_float4(r0.x, r1.x, r0.y, r1.y);
    *(float4*)(dp + 4) = make_float4(r0.z, r1.z, r0.w, r1.w);
  }
  __syncthreads();

  const int wave = (int)((blockIdx.x * blockDim.x + threadIdx.x) >> 5);
  const int lane = (int)(threadIdx.x & 31);
  if (wave * 32 >= M) return;           // wave-uniform; after the barrier
  const int half = lane >> 4;           // 0: K pair {k0,k0+1}, 1: K pair {k0+2,k0+3}
  const int l15  = lane & 15;
  const int kk   = half * 2;
  const float* Arow0 = A + (size_t)(wave * 32 + l15) * K + kk;
  const float* Arow1 = Arow0 + (size_t)16 * K;
  const float* bbase = sW + half * PSTR + 2 * l15;   // this lane's fragment base

  v8f acc[16];
#pragma unroll
  for (int i = 0; i < 16; ++i)
#pragma unroll
    for (int j = 0; j < 8; ++j) acc[i][j] = 0.0f;

  constexpr int KMAIN = K & ~3;         // 52 or 128
  for (int k0 = 0; k0 < KMAIN; k0 += 4) {
    const v2f a0 = *(const v2f*)(Arow0 + k0);
    const v2f a1 = *(const v2f*)(Arow1 + k0);
    const float* bp = bbase + (k0 >> 1) * PSTR;
#pragma unroll
    for (int nt = 0; nt < 8; ++nt) {
      const v2f b = *(const v2f*)(bp + nt * 32);     // one ds_load_b64 per fragment
      acc[nt] = __builtin_amdgcn_wmma_f32_16x16x4_f32(
          false, a0, false, b, (short)0, acc[nt], false, false);
      acc[8 + nt] = __builtin_amdgcn_wmma_f32_16x16x4_f32(
          false, a1, false, b, (short)0, acc[8 + nt], false, false);
    }
  }
  if constexpr (KMAIN != K) {           // K=54 tail: rows 52..55 (54,55 staged as zero)
    const int ka = KMAIN + kk;
    const int kc = (ka < K) ? ka : (K - 2);          // clamped, unconditional load
    const float msk = (ka < K) ? 1.0f : 0.0f;
    const float* A0 = A + (size_t)(wave * 32 + l15) * K + kc;
    const float* A1 = A0 + (size_t)16 * K;
    v2f a0 = *(const v2f*)A0;  a0[0] *= msk; a0[1] *= msk;
    v2f a1 = *(const v2f*)A1;  a1[0] *= msk; a1[1] *= msk;
    const float* bp = bbase + (KMAIN >> 1) * PSTR;
#pragma unroll
    for (int nt = 0; nt < 8; ++nt) {
      const v2f b = *(const v2f*)(bp + nt * 32);
      acc[nt] = __builtin_amdgcn_wmma_f32_16x16x4_f32(
          false, a0, false, b, (short)0, acc[nt], false, false);
      acc[8 + nt] = __builtin_amdgcn_wmma_f32_16x16x4_f32(
          false, a1, false, b, (short)0, acc[8 + nt], false, false);
    }
  }

  // C/D layout: VGPR r -> row base+r (lanes 0-15) / base+8+r (lanes 16-31)
#pragma unroll
  for (int set = 0; set < 2; ++set) {
    const int mbase = wave * 32 + set * 16 + half * 8;
#pragma unroll
    for (int nt = 0; nt < 8; ++nt) {
      const int col = nt * 16 + l15;
      const float bv = (flags & 1) ? bias[col] : 0.0f;
#pragma unroll
      for (int r = 0; r < 8; ++r) {
        float v = acc[set * 8 + nt][r] + bv;
        if (flags & 2) v = fmaxf(v, 0.0f);
        if (flags & 4) v += resid[(size_t)(mbase + r) * DGD + col];
        C[(size_t)(mbase + r) * DGD + col] = v;
      }
    }
  }
}

// ---------- GAT: edge attention logits + running max (one wave per edge) ----------
__global__ __launch_bounds__(256) void k_gat_logits(
    const float* __restrict__ hh, const float* __restrict__ hA,
    const int* __restrict__ ei, float* __restrict__ elog, unsigned* __restrict__ nmax) {
  const int w = (int)((blockIdx.x * blockDim.x + threadIdx.x) >> 5);
  const int lane = (int)(threadIdx.x & 31);
  if (w >= NE1) return;
  const int src = ei[w], dst = ei[NE1 + w];
  const float4 ts = *(const float4*)(hh + (size_t)src * DGD + lane * 4);
  const float4 td = *(const float4*)(hh + (size_t)dst * DGD + lane * 4);
  const float4 as = *(const float4*)(hA + (size_t)src * DGD + lane * 4);
  const float4 ad = *(const float4*)(hA + (size_t)dst * DGD + lane * 4);
  float p = ad.x * ts.x + ad.y * ts.y + ad.z * ts.z + ad.w * ts.w
          + as.x * td.x + as.y * td.y + as.z * td.z + as.w * td.w;
  p = wred(p);
  if (lane == 0) { elog[w] = p; atomicMax(&nmax[dst], fkey(p)); }
}

// ---------- GAT: exp + segment sum (one thread per edge) ----------
__global__ __launch_bounds__(256) void k_gat_exp(
    const int* __restrict__ ei, const unsigned* __restrict__ nmax,
    float* __restrict__ elog, float* __restrict__ nsum) {
  const int e = (int)(blockIdx.x * blockDim.x + threadIdx.x);
  if (e >= NE1) return;
  const int dst = ei[NE1 + e];
  const float m = funkey(nmax[dst]);
  const float ex = __expf(elog[e] - m);
  elog[e] = ex;
  atomicAdd(&nsum[dst], ex);
}

// ---------- GAT: weighted scatter of h[src] into hp[dst] (one wave per edge) ----------
__global__ __launch_bounds__(256) void k_gat_accum(
    const float* __restrict__ hh, const int* __restrict__ ei,
    const float* __restrict__ elog, const float* __restrict__ nsum,
    float* __restrict__ hp) {
  const int w = (int)((blockIdx.x * blockDim.x + threadIdx.x) >> 5);
  const int lane = (int)(threadIdx.x & 31);
  if (w >= NE1) return;
  const int src = ei[w], dst = ei[NE1 + w];
  const float att = elog[w] / (nsum[dst] + 1e-16f);
  const float4 ts = *(const float4*)(hh + (size_t)src * DGD + lane * 4);
  float* d = hp + (size_t)dst * DGD + lane * 4;
  atomicAdd(d + 0, att * ts.x);
  atomicAdd(d + 1, att * ts.y);
  atomicAdd(d + 2, att * ts.z);
  atomicAdd(d + 3, att * ts.w);
}

// ---------- GAT: gated update, in place (one wave per node) ----------
__global__ __launch_bounds__(256) void k_gat_update(
    float* __restrict__ H, const float* __restrict__ hpacc,
    const float* __restrict__ gW, const float* __restrict__ gb) {
  const int n = (int)((blockIdx.x * blockDim.x + threadIdx.x) >> 5);
  const int lane = (int)(threadIdx.x & 31);
  if (n >= NN) return;
  float4 x4 = *(const float4*)(H + (size_t)n * DGD + lane * 4);
  float4 p4 = *(const float4*)(hpacc + (size_t)n * DGD + lane * 4);
  p4.x = fmaxf(p4.x, 0.f); p4.y = fmaxf(p4.y, 0.f);
  p4.z = fmaxf(p4.z, 0.f); p4.w = fmaxf(p4.w, 0.f);
  const float4 g0 = *(const float4*)(gW + lane * 4);
  const float4 g1 = *(const float4*)(gW + DGD + lane * 4);
  float s = x4.x * g0.x + x4.y * g0.y + x4.z * g0.z + x4.w * g0.w
          + p4.x * g1.x + p4.y * g1.y + p4.z * g1.z + p4.w * g1.w;
  s = wred(s);
  const float c = 1.0f / (1.0f + __expf(-(s + gb[0])));
  float4 o;
  o.x = c * x4.x + (1.f - c) * p4.x;
  o.y = c * x4.y + (1.f - c) * p4.y;
  o.z = c * x4.z + (1.f - c) * p4.z;
  o.w = c * x4.w + (1.f - c) * p4.w;
  *(float4*)(H + (size_t)n * DGD + lane * 4) = o;
}

// ---------- InterNet scatter: m[dst] += mproj[src] (one wave per edge) ----------
__global__ __launch_bounds__(256) void k_scatter(
    const float* __restrict__ T, const int* __restrict__ ei, float* __restrict__ U) {
  const int w = (int)((blockIdx.x * blockDim.x + threadIdx.x) >> 5);
  const int lane = (int)(threadIdx.x & 31);
  if (w >= NE2) return;
  const int src = ei[w], dst = ei[NE2 + w];
  const float4 t = *(const float4*)(T + (size_t)src * DGD + lane * 4);
  float* d = U + (size_t)dst * DGD + lane * 4;
  atomicAdd(d + 0, t.x);
  atomicAdd(d + 1, t.y);
  atomicAdd(d + 2, t.z);
  atomicAdd(d + 3, t.w);
}

// ---------- interaction physics: one wave per 8 consecutive edges ----------
__global__ __launch_bounds__(256) void k_interact(
    const float* __restrict__ Tv0, const float* __restrict__ Tv1,
    const float* __restrict__ Td0, const float* __restrict__ Td1,
    const float* __restrict__ w2v, const float* __restrict__ b2v,
    const float* __restrict__ w2d, const float* __restrict__ b2d,
    const float* __restrict__ pos, const float* __restrict__ vdw,
    const int* __restrict__ ei, const unsigned char* __restrict__ fmet,
    const unsigned char* __restrict__ fdon, const unsigned char* __restrict__ facc,
    const unsigned char* __restrict__ fhyd, const int* __restrict__ batch,
    const float* __restrict__ hbond_c, const float* __restrict__ hydro_c,
    float* __restrict__ eacc, float* __restrict__ out_dvdw) {
  const int wave = (int)((blockIdx.x * blockDim.x + threadIdx.x) >> 5);
  const int lane = (int)(threadIdx.x & 31);
  const int e0 = wave * 8;
  if (e0 >= NE3) return;
  const float4 gv = *(const float4*)(w2v + lane * 4);
  const float4 gd = *(const float4*)(w2d + lane * 4);
  const float bbv = b2v[0], bbd = b2d[0];
  const float hbc = hbond_c[0], hyc = hydro_c[0];
  const float min_hb = -(hbc * hbc), min_hy = -(hyc * hyc);
  float s0 = 0.f, s1 = 0.f, s2 = 0.f, s3 = 0.f;
  int g = 0;
  for (int q = 0; q < 8; ++q) {
    const int e = e0 + q;
    const int i0 = ei[e], i1 = ei[NE3 + e];
    const float4 av = *(const float4*)(Tv0 + (size_t)i0 * DGD + lane * 4);
    const float4 bv = *(const float4*)(Tv1 + (size_t)i1 * DGD + lane * 4);
    const float4 ad = *(const float4*)(Td0 + (size_t)i0 * DGD + lane * 4);
    const float4 bd = *(const float4*)(Td1 + (size_t)i1 * DGD + lane * 4);
    float pe = fmaxf(av.x + bv.x, 0.f) * gv.x + fmaxf(av.y + bv.y, 0.f) * gv.y
             + fmaxf(av.z + bv.z, 0.f) * gv.z + fmaxf(av.w + bv.w, 0.f) * gv.w;
    float pd = fmaxf(ad.x + bd.x, 0.f) * gd.x + fmaxf(ad.y + bd.y, 0.f) * gd.y
             + fmaxf(ad.z + bd.z, 0.f) * gd.z + fmaxf(ad.w + bd.w, 0.f) * gd.w;
#pragma unroll
    for (int off = 16; off > 0; off >>= 1) {
      pe += __shfl_xor(pe, off, 32);
      pd += __shfl_xor(pd, off, 32);
    }
    float eps = 1.0f / (1.0f + __expf(-(pe + bbv)));
    eps = eps * (0.0356f - 0.0178f) + 0.0178f;
    const float dv = tanhf(pd + bbd) * 0.2f;
    const float dx = pos[(size_t)i0 * 3 + 0] - pos[(size_t)i1 * 3 + 0];
    const float dy = pos[(size_t)i0 * 3 + 1] - pos[(size_t)i1 * 3 + 1];
    const float dz = pos[(size_t)i0 * 3 + 2] - pos[(size_t)i1 * 3 + 2];
    const float D = sqrtf(dx * dx + dy * dy + dz * dz + 1e-12f);
    const float maskd = (D >= 0.5f && D <= 5.0f) ? 1.f : 0.f;
    const float Dc = fminf(fmaxf(D, 0.5f), 5.0f);
    const float R = vdw[i0] + vdw[i1] + dv;
    const float rr = R / Dc;
    const float rr2 = rr * rr;
    const float p6 = rr2 * rr2 * rr2;
    const float p10 = p6 * rr2 * rr2;
    const float p = (Dc > R) ? p6 : p10;
    const float evdw = fminf(eps * (p * p - 2.f * p), 100.f);
    const float dmr = Dc - R;
    const float fhb = fminf(fmaxf((0.0f - dmr) / 0.7f, 0.f), 1.f);
    const float fhy = fminf(fmaxf((1.5f - dmr) / 1.0f, 0.f), 1.f);
    const bool mi = fmet[i0], mj = fmet[i1];
    const bool di = fdon[i0], dj = fdon[i1];
    const bool ai = facc[i0], aj = facc[i1];
    const bool yi = fhyd[i0], yj = fhyd[i1];
    const float mhb = (((di && aj) || (ai && dj)) && !(mi || mj)) ? 1.f : 0.f;
    const float mme = ((mi && (dj || aj)) || (mj && (di || ai))) ? 1.f : 0.f;
    const float mhy = (yi && yj) ? 1.f : 0.f;
    s0 += evdw * maskd;
    s1 += (min_hb * fhb) * mhb * maskd;
    s2 += (min_hb * fhb) * mme * maskd;
    s3 += (min_hy * fhy) * mhy * maskd;
    if (lane == 0) out_dvdw[e] = dv * maskd;
    g = batch[i0];
  }
  if (lane == 0) {
    atomicAdd(&eacc[g * 4 + 0], s0);
    atomicAdd(&eacc[g * 4 + 1], s1);
    atomicAdd(&eacc[g * 4 + 2], s2);
    atomicAdd(&eacc[g * 4 + 3], s3);
  }
}

// ---------- finalize: divide by rotor penalty ----------
__global__ void k_final(const float* __restrict__ eacc, const float* __restrict__ rotor,
                        const float* __restrict__ rotor_c, float* __restrict__ out) {
  const int t = (int)(blockIdx.x * blockDim.x + threadIdx.x);
  if (t >= NBATCH * 4) return;
  const float rc = rotor_c[0];
  out[t] = eacc[t] / (1.0f + rc * rc * rotor[t >> 2]);
}

extern "C" void kernel_launch(void* const* d_in, const int* in_sizes, int n_in,
                              void* d_out, int out_size, void* d_ws, size_t ws_size,
                              hipStream_t stream) {
  (void)in_sizes; (void)n_in; (void)out_size; (void)ws_size;
  // ---- inputs (setup_inputs insertion order; params flattened in insertion order) ----
  const float* x      = (const float*)d_in[0];
  const float* pos    = (const float*)d_in[1];
  const float* vdwr   = (const float*)d_in[2];
  const float* rotor  = (const float*)d_in[3];
  const float* embedW = (const float*)d_in[4];
  // per-layer params: base 5 + 9*l :
  //  +0 gat_W, +1 gat_b, +2 gat_A, +3 gat_gW, +4 gat_gb, +5 int_Wm, +6 int_bm, +7 int_Wu, +8 int_bu
  const float* vepsW1 = (const float*)d_in[32];
  const float* vepsB1 = (const float*)d_in[33];
  const float* vepsW2 = (const float*)d_in[34];
  const float* vepsB2 = (const float*)d_in[35];
  const float* dvdwW1 = (const float*)d_in[36];
  const float* dvdwB1 = (const float*)d_in[37];
  const float* dvdwW2 = (const float*)d_in[38];
  const float* dvdwB2 = (const float*)d_in[39];
  const float* hbondC = (const float*)d_in[40];
  const float* hydroC = (const float*)d_in[41];
  const float* rotorC = (const float*)d_in[42];
  const int* ei    = (const int*)d_in[43];
  const int* eic   = (const int*)d_in[44];
  const int* eii   = (const int*)d_in[45];
  const int* batch = (const int*)d_in[46];
  const unsigned char* fmet = (const unsigned char*)d_in[47];
  const unsigned char* fdon = (const unsigned char*)d_in[48];
  const unsigned char* facc = (const unsigned char*)d_in[49];
  const unsigned char* fhyd = (const unsigned char*)d_in[50];
  float* out = (float*)d_out;            // [96*4 energies][576000 dvdw]

  // ---- workspace layout (floats) ----
  const size_t NBf = (size_t)NN * DGD;   // 2,826,240 floats per node buffer
  float* H   = (float*)d_ws;
  float* T   = H + NBf;
  float* U   = H + 2 * NBf;
  float* X0  = H + 3 * NBf;
  float* X1  = H + 4 * NBf;
  float* Eb  = H + 5 * NBf;              // NE1 floats (edge logits / exp)
  unsigned* nmax = (unsigned*)(Eb + NE1);
  float* nsum = (float*)(nmax + NN);
  float* eacc = nsum + NN;               // 384 floats

  const dim3 blk(256);
  const int gemmGrid  = (NN / 32 + 7) / 8;   // 690 waves (32 rows each) -> 87 blocks
  const int edge1Grid = NE1 / 8;             // wave per edge
  const int edge1TGrid = (NE1 + 255) / 256;  // thread per edge
  const int edge2Grid = NE2 / 8;
  const int nodeGrid  = NN / 8;
  const int intGrid   = (NE3 / 8 + 7) / 8;   // 72000 waves -> 9000 blocks

  // ---- embed: H = x @ embed_W (K = 54) ----
  k_gemm_wmma<FIND><<<gemmGrid, blk, 0, stream>>>(x, embedW, nullptr, nullptr, H, NN, 0);

  // ---- 3 gated-GAT layers on intra edges ----
  for (int l = 0; l < 3; ++l) {
    const float* gW  = (const float*)d_in[5 + 9 * l + 0];
    const float* gB  = (const float*)d_in[5 + 9 * l + 1];
    const float* gA  = (const float*)d_in[5 + 9 * l + 2];
    const float* ggW = (const float*)d_in[5 + 9 * l + 3];
    const float* ggB = (const float*)d_in[5 + 9 * l + 4];
    k_gemm_wmma<DGD><<<gemmGrid, blk, 0, stream>>>(H, gW, gB, nullptr, T, NN, 1);      // T = H@W+b
    k_gemm_wmma<DGD><<<gemmGrid, blk, 0, stream>>>(T, gA, nullptr, nullptr, U, NN, 0); // U = T@A
    hipMemsetAsync(X0, 0, NBf * sizeof(float), stream);
    hipMemsetAsync(nmax, 0, NN * sizeof(unsigned), stream);
    hipMemsetAsync(nsum, 0, NN * sizeof(float), stream);
    k_gat_logits<<<edge1Grid, blk, 0, stream>>>(T, U, ei, Eb, nmax);
    k_gat_exp<<<edge1TGrid, blk, 0, stream>>>(ei, nmax, Eb, nsum);
    k_gat_accum<<<edge1Grid, blk, 0, stream>>>(T, ei, Eb, nsum, X0);
    k_gat_update<<<nodeGrid, blk, 0, stream>>>(H, X0, ggW, ggB);
  }

  // ---- 3 InterNet layers on conv edges ----
  for (int l = 0; l < 3; ++l) {
    const float* Wm = (const float*)d_in[5 + 9 * l + 5];
    const float* bm = (const float*)d_in[5 + 9 * l + 6];
    const float* Wu = (const float*)d_in[5 + 9 * l + 7];
    const float* bu = (const float*)d_in[5 + 9 * l + 8];
    k_gemm_wmma<DGD><<<gemmGrid, blk, 0, stream>>>(H, Wm, bm, nullptr, T, NN, 1 | 2);
    hipMemsetAsync(U, 0, NBf * sizeof(float), stream);
    k_scatter<<<edge2Grid, blk, 0, stream>>>(T, eic, U);
    k_gemm_wmma<DGD><<<gemmGrid, blk, 0, stream>>>(U, Wu, bu, H, H, NN, 1 | 2 | 4);
  }

  // ---- per-node interaction projections (factor xc@W1 across the concat) ----
  k_gemm_wmma<DGD><<<gemmGrid, blk, 0, stream>>>(H, vepsW1, nullptr, nullptr, T, NN, 0);
  k_gemm_wmma<DGD><<<gemmGrid, blk, 0, stream>>>(H, vepsW1 + 128 * 128, vepsB1, nullptr, U, NN, 1);
  k_gemm_wmma<DGD><<<gemmGrid, blk, 0, stream>>>(H, dvdwW1, nullptr, nullptr, X0, NN, 0);
  k_gemm_wmma<DGD><<<gemmGrid, blk, 0, stream>>>(H, dvdwW1 + 128 * 128, dvdwB1, nullptr, X1, NN, 1);

  // ---- physics over all ligand x pocket pairs ----
  hipMemsetAsync(eacc, 0, NBATCH * 4 * sizeof(float), stream);
  k_interact<<<intGrid, blk, 0, stream>>>(T, U, X0, X1, vepsW2, vepsB2, dvdwW2, dvdwB2,
                                          pos, vdwr, eii, fmet, fdon, facc, fhyd, batch,
                                          hbondC, hydroC, eacc, out + NBATCH * 4);
  k_final<<<2, blk, 0, stream>>>(eacc, rotor, rotorC, out);
}